// LossFunction_62981400429247
// MI455X (gfx1250) — compile-verified
//
#include <hip/hip_runtime.h>
#include <hip/hip_bf16.h>
#include <math.h>

typedef __attribute__((ext_vector_type(2))) float v2f;
typedef __attribute__((ext_vector_type(8))) float v8f;

#define DIMS 128
#define KSTEPS 32           // DIMS / 4
#define CSPLIT 8            // column chunks per row-block
#define EPS_F 1e-5f
#define THRESH_F 0.5f

// ---------------------------------------------------------------------------
// Pass 1: partial per-row min positive / max negative sim over a column chunk.
// One wave owns (rowblk, chunk). f32 WMMA C layout: acc[i], lanes 0-15 ->
// row i, lanes 16-31 -> row i+8, col = lane & 15.
// ---------------------------------------------------------------------------
__global__ __launch_bounds__(256) void ms_pass1(
    const float* __restrict__ feats, const int* __restrict__ labels2,
    float* __restrict__ mnP, float* __restrict__ mxP, int nrows)
{
    const int gwave   = blockIdx.x * 8 + (threadIdx.x >> 5);
    const int lane    = threadIdx.x & 31;
    const int lane_lo = lane & 15;
    const int half    = lane >> 4;

    const int rowblk   = gwave >> 3;          // / CSPLIT
    const int chunk    = gwave & (CSPLIT - 1);
    const int row_base = rowblk * 16;
    if (row_base >= nrows) return;            // wave-uniform exit

    const int ccols     = nrows / CSPLIT;
    const int col_begin = chunk * ccols;
    const int col_end   = col_begin + ccols;

    // Preload A tile (16x128) once: 64 VGPRs.
    const float* arow = feats + (size_t)(row_base + lane_lo) * DIMS + 2 * half;
    v2f a[KSTEPS];
#pragma unroll
    for (int k = 0; k < KSTEPS; ++k)
        a[k] = *(const v2f*)(arow + 4 * k);

    int rl[8];
#pragma unroll
    for (int i = 0; i < 8; ++i)
        rl[i] = labels2[2 * (row_base + i + 8 * half)];   // low dword of int64

    float mn[8], mx[8];
#pragma unroll
    for (int i = 0; i < 8; ++i) { mn[i] = INFINITY; mx[i] = -INFINITY; }

    const float* bbase = feats + (size_t)lane_lo * DIMS + 2 * half;

    for (int cb = col_begin; cb < col_end; cb += 16) {
        const float* brow = bbase + (size_t)cb * DIMS;
        v2f b[KSTEPS];
#pragma unroll
        for (int k = 0; k < KSTEPS; ++k)
            b[k] = *(const v2f*)(brow + 4 * k);
        const int cl = labels2[2 * (cb + lane_lo)];
        v8f c = {0.f, 0.f, 0.f, 0.f, 0.f, 0.f, 0.f, 0.f};
#pragma unroll
        for (int k = 0; k < KSTEPS; ++k)
            c = __builtin_amdgcn_wmma_f32_16x16x4_f32(
                    false, a[k], false, b[k], (short)0, c, false, false);
#pragma unroll
        for (int i = 0; i < 8; ++i) {
            const float s   = c[i];
            const bool same = (rl[i] == cl);
            mn[i] = (same && s < 1.0f - EPS_F) ? fminf(mn[i], s) : mn[i];
            mx[i] = (!same)                    ? fmaxf(mx[i], s) : mx[i];
        }
    }

    // Reduce across the 16 lanes of each half (xor masks never cross bit 4).
#pragma unroll
    for (int i = 0; i < 8; ++i) {
#pragma unroll
        for (int m = 1; m < 16; m <<= 1) {
            mn[i] = fminf(mn[i], __shfl_xor(mn[i], m, 32));
            mx[i] = fmaxf(mx[i], __shfl_xor(mx[i], m, 32));
        }
    }
    if (lane_lo == 0) {
#pragma unroll
        for (int i = 0; i < 8; ++i) {
            const int r = row_base + i + 8 * half;
            mnP[chunk * nrows + r] = mn[i];
            mxP[chunk * nrows + r] = mx[i];
        }
    }
}

// ---------------------------------------------------------------------------
// Combine pass-1 chunk partials (fixed order -> deterministic).
// ---------------------------------------------------------------------------
__global__ __launch_bounds__(256) void ms_combine1(
    const float* __restrict__ mnP, const float* __restrict__ mxP,
    float* __restrict__ minpos, float* __restrict__ maxneg, int nrows)
{
    const int r = blockIdx.x * 256 + threadIdx.x;
    if (r >= nrows) return;
    float mn = INFINITY, mx = -INFINITY;
#pragma unroll
    for (int c = 0; c < CSPLIT; ++c) {
        mn = fminf(mn, mnP[c * nrows + r]);
        mx = fmaxf(mx, mxP[c * nrows + r]);
    }
    minpos[r] = mn;
    maxneg[r] = mx;
}

// ---------------------------------------------------------------------------
// Pass 2: recompute sim over column chunk; partial masked exp-sums + flags.
// ---------------------------------------------------------------------------
__global__ __launch_bounds__(256) void ms_pass2(
    const float* __restrict__ feats, const int* __restrict__ labels2,
    const float* __restrict__ minpos, const float* __restrict__ maxneg,
    const float* __restrict__ p_margin, const float* __restrict__ p_sp,
    const float* __restrict__ p_sn,
    float* __restrict__ psumP, float* __restrict__ nsumP,
    float* __restrict__ hpP, float* __restrict__ hnP, int nrows)
{
    const int gwave   = blockIdx.x * 8 + (threadIdx.x >> 5);
    const int lane    = threadIdx.x & 31;
    const int lane_lo = lane & 15;
    const int half    = lane >> 4;

    const int rowblk   = gwave >> 3;
    const int chunk    = gwave & (CSPLIT - 1);
    const int row_base = rowblk * 16;
    if (row_base >= nrows) return;

    const int ccols     = nrows / CSPLIT;
    const int col_begin = chunk * ccols;
    const int col_end   = col_begin + ccols;

    const float margin = *p_margin;
    const float sp     = *p_sp;
    const float sn     = *p_sn;

    const float* arow = feats + (size_t)(row_base + lane_lo) * DIMS + 2 * half;
    v2f a[KSTEPS];
#pragma unroll
    for (int k = 0; k < KSTEPS; ++k)
        a[k] = *(const v2f*)(arow + 4 * k);

    int   rl[8];
    float nth[8];   // min_pos - margin  (neg-pair threshold)
    float pth[8];   // max_neg + margin  (pos-pair threshold)
#pragma unroll
    for (int i = 0; i < 8; ++i) {
        const int r = row_base + i + 8 * half;
        rl[i]  = labels2[2 * r];
        nth[i] = minpos[r] - margin;
        pth[i] = maxneg[r] + margin;
    }

    float psum[8], nsum[8], hp[8], hn[8];
#pragma unroll
    for (int i = 0; i < 8; ++i) { psum[i] = 0.f; nsum[i] = 0.f; hp[i] = 0.f; hn[i] = 0.f; }

    const float* bbase = feats + (size_t)lane_lo * DIMS + 2 * half;

    for (int cb = col_begin; cb < col_end; cb += 16) {
        const float* brow = bbase + (size_t)cb * DIMS;
        v2f b[KSTEPS];
#pragma unroll
        for (int k = 0; k < KSTEPS; ++k)
            b[k] = *(const v2f*)(brow + 4 * k);
        const int cl = labels2[2 * (cb + lane_lo)];
        v8f c = {0.f, 0.f, 0.f, 0.f, 0.f, 0.f, 0.f, 0.f};
#pragma unroll
        for (int k = 0; k < KSTEPS; ++k)
            c = __builtin_amdgcn_wmma_f32_16x16x4_f32(
                    false, a[k], false, b[k], (short)0, c, false, false);
#pragma unroll
        for (int i = 0; i < 8; ++i) {
            const float s   = c[i];
            const bool same = (rl[i] == cl);
            const bool nm   = (!same) && (s > nth[i]);
            const bool pm   = same && (s < 1.0f - EPS_F) && (s < pth[i]);
            nsum[i] += nm ? __expf( sn * (s - THRESH_F)) : 0.f;
            psum[i] += pm ? __expf(-sp * (s - THRESH_F)) : 0.f;
            hn[i]    = nm ? 1.f : hn[i];
            hp[i]    = pm ? 1.f : hp[i];
        }
    }

#pragma unroll
    for (int i = 0; i < 8; ++i) {
#pragma unroll
        for (int m = 1; m < 16; m <<= 1) {
            psum[i] += __shfl_xor(psum[i], m, 32);
            nsum[i] += __shfl_xor(nsum[i], m, 32);
            hp[i]    = fmaxf(hp[i], __shfl_xor(hp[i], m, 32));
            hn[i]    = fmaxf(hn[i], __shfl_xor(hn[i], m, 32));
        }
    }

    if (lane_lo == 0) {
#pragma unroll
        for (int i = 0; i < 8; ++i) {
            const int r = row_base + i + 8 * half;
            psumP[chunk * nrows + r] = psum[i];
            nsumP[chunk * nrows + r] = nsum[i];
            hpP  [chunk * nrows + r] = hp[i];
            hnP  [chunk * nrows + r] = hn[i];
        }
    }
}

// ---------------------------------------------------------------------------
// Finalize: fixed-order chunk combine + per-row loss + deterministic LDS tree
// reduction; single block writes out[0..2) directly (no atomics, no pre-zero).
// ---------------------------------------------------------------------------
__global__ __launch_bounds__(1024) void ms_finalize(
    const float* __restrict__ psumP, const float* __restrict__ nsumP,
    const float* __restrict__ hpP, const float* __restrict__ hnP,
    const float* __restrict__ p_sp, const float* __restrict__ p_sn,
    float* __restrict__ out, int nrows)
{
    __shared__ float s0[1024];
    __shared__ float s1[1024];
    const int tid = threadIdx.x;
    const float sp = *p_sp;
    const float sn = *p_sn;

    float lsum = 0.f, cnt = 0.f;
    for (int r = tid; r < nrows; r += 1024) {
        float ps = 0.f, ns = 0.f, hp = 0.f, hn = 0.f;
#pragma unroll
        for (int c = 0; c < CSPLIT; ++c) {
            ps += psumP[c * nrows + r];
            ns += nsumP[c * nrows + r];
            hp  = fmaxf(hp, hpP[c * nrows + r]);
            hn  = fmaxf(hn, hnP[c * nrows + r]);
        }
        const bool hasN = hn > 0.5f;
        const bool hasP = hp > 0.5f;
        if (hasN && hasP)
            lsum += log1pf(ps) / sp + log1pf(ns) / sn;
        if (!hasN) cnt += 1.f;
    }
    s0[tid] = lsum;
    s1[tid] = cnt;
    __syncthreads();
    for (int o = 512; o > 0; o >>= 1) {
        if (tid < o) { s0[tid] += s0[tid + o]; s1[tid] += s1[tid + o]; }
        __syncthreads();
    }
    if (tid == 0) {
        out[0] = s0[0] / (float)nrows;
        out[1] = s1[0] / (float)nrows;
    }
}

extern "C" void kernel_launch(void* const* d_in, const int* in_sizes, int n_in,
                              void* d_out, int out_size, void* d_ws, size_t ws_size,
                              hipStream_t stream) {
    const float* feats   = (const float*)d_in[0];
    const int*   labels2 = (const int*)d_in[1];     // int64 viewed as int pairs (LE)
    const float* margin  = (const float*)d_in[2];
    const float* sp      = (const float*)d_in[3];
    const float* sn      = (const float*)d_in[4];
    float*       out     = (float*)d_out;

    const int nrows = in_sizes[1];                  // B = 8192
    const size_t R  = (size_t)nrows;

    float* minpos = (float*)d_ws;                   // [R]
    float* maxneg = minpos + R;                     // [R]
    float* mnP    = maxneg + R;                     // [CSPLIT*R]
    float* mxP    = mnP + CSPLIT * R;               // [CSPLIT*R]
    float* psumP  = mxP + CSPLIT * R;               // [CSPLIT*R]
    float* nsumP  = psumP + CSPLIT * R;             // [CSPLIT*R]
    float* hpP    = nsumP + CSPLIT * R;             // [CSPLIT*R]
    float* hnP    = hpP + CSPLIT * R;               // [CSPLIT*R]

    const int rowblks = (nrows + 15) / 16;          // 512
    const int grid    = rowblks;                    // 512 blocks x 8 waves = 4096 waves

    ms_pass1<<<grid, 256, 0, stream>>>(feats, labels2, mnP, mxP, nrows);
    ms_combine1<<<(nrows + 255) / 256, 256, 0, stream>>>(mnP, mxP, minpos, maxneg, nrows);
    ms_pass2<<<grid, 256, 0, stream>>>(feats, labels2, minpos, maxneg,
                                       margin, sp, sn, psumP, nsumP, hpP, hnP, nrows);
    ms_finalize<<<1, 1024, 0, stream>>>(psumP, nsumP, hpP, hnP, sp, sn, out, nrows);
}